// GraphSSM_53360673686108
// MI455X (gfx1250) — compile-verified
//
#include <hip/hip_runtime.h>
#include <math.h>

// ---------------------------------------------------------------------------
// GraphSSM for MI455X (gfx1250, wave32).
//
// Heavy math runs on V_WMMA_F32_16X16X4_F32 (f32 matrix pipe). Eigvec row
// normalization, BatchNorm and SiLU are folded into the A-tile loads.
// B tiles (pure copies) use GLOBAL_LOAD_ASYNC_TO_LDS_B128 when the toolchain
// exposes the builtin (ASYNCcnt-tracked, completed with s_wait_asynccnt),
// else vectorized b128 loads + LDS stores. The complex SSM core collapses to:
//   R[b,k,d] = QTX[b,k,d] * sum_s G(ev[b,k], Abar[d,s]) * (delta*Bp*Cp)[d,s]
// ---------------------------------------------------------------------------

typedef __attribute__((ext_vector_type(2))) float v2f;
typedef __attribute__((ext_vector_type(8))) float v8f;
typedef int v4i_vs __attribute__((__vector_size__(4 * sizeof(int))));

#define MT 128              // block tile M
#define NT 64               // block tile N
#define KT 32               // block tile K (reduction)
#define A_STRIDE (MT + 4)   // 132: multiple of 4 -> 16B-aligned float4 rows
#define B_STRIDE (NT + 4)   // 68

#if defined(__has_builtin)
#if __has_builtin(__builtin_amdgcn_global_load_async_to_lds_b128)
#define USE_ASYNC_LDS 1
#endif
#endif
#ifndef USE_ASYNC_LDS
#define USE_ASYNC_LDS 0
#endif

// Typed address-space casts matching the builtin's signature:
// param0: v4i in AS1 (global, printed by clang as "__device__"),
// param1: v4i in AS3 (LDS).
#define AS1CAST(p) ((__attribute__((address_space(1))) v4i_vs*)(unsigned long long)(p))
#define AS3CAST(p) ((__attribute__((address_space(3))) v4i_vs*)(unsigned long long)(p))

__device__ __forceinline__ void wait_async_lds() {
#if USE_ASYNC_LDS
#if __has_builtin(__builtin_amdgcn_s_wait_asynccnt)
    __builtin_amdgcn_s_wait_asynccnt(0);
#else
    asm volatile("s_wait_asynccnt 0x0" ::: "memory");
#endif
#endif
}

__device__ __forceinline__ float nan0(float v) { return (v == v) ? v : 0.0f; }
__device__ __forceinline__ float silu(float v) { return v / (1.0f + __expf(-v)); }

// AMODE: 0 = plain, 1 = nan0(a) * scale[row_L], 2 = batchnorm over reduction idx,
//        3 = silu(a).   TRANSA: A'[m][r] = A[r*lda+m] (only used with AMODE 1).
template <int AMODE, bool TRANSA, bool HAS_BIAS>
__global__ __launch_bounds__(256) void wmma_gemm(
    const float* __restrict__ Ag, const float* __restrict__ Bg,
    float* __restrict__ Cg,
    const float* __restrict__ p0, const float* __restrict__ p1,
    const float* __restrict__ p2, const float* __restrict__ bias,
    int M, int N, int R, int lda,
    long long strideA, long long strideB, long long strideC, long long strideP)
{
    __shared__ float ldsA[KT][A_STRIDE];
    __shared__ float ldsB[KT][B_STRIDE];

    const int b = blockIdx.z;
    const float* A = Ag + (long long)b * strideA;
    const float* B = Bg + (long long)b * strideB;
    float*       C = Cg + (long long)b * strideC;
    const float* scl = p0 + (long long)b * strideP;   // AMODE 1 only

    const int m0   = blockIdx.x * MT;
    const int n0   = blockIdx.y * NT;
    const int tid  = threadIdx.x;
    const int lane = tid & 31;
    const int wave = tid >> 5;
    const int wm   = (wave & 3) * 32;   // wave M offset inside block tile
    const int wn   = (wave >> 2) * 32;  // wave N offset inside block tile
    const int half = lane >> 4;         // 0: K pair {0,1}, 1: K pair {2,3}
    const int lrow = lane & 15;

    v8f acc[2][2];
    const v8f vzero = {0.f,0.f,0.f,0.f,0.f,0.f,0.f,0.f};
    acc[0][0] = vzero; acc[0][1] = vzero; acc[1][0] = vzero; acc[1][1] = vzero;

    for (int r0 = 0; r0 < R; r0 += KT) {
        // ---- B tile (KT x NT): pure copy global -> LDS ----
        // 512 float4 across 256 threads -> 2 each.
        #pragma unroll
        for (int t = 0; t < (KT * NT) / (256 * 4); ++t) {
            int idx = tid + t * 256;
            int i  = idx >> 4;          // r offset (0..31)
            int j4 = idx & 15;          // float4 column
            const float* gsrc = &B[(long long)(r0 + i) * N + (n0 + j4 * 4)];
            float* ldst = &ldsB[i][j4 * 4];
#if USE_ASYNC_LDS
            __builtin_amdgcn_global_load_async_to_lds_b128(AS1CAST(gsrc), AS3CAST(ldst), 0, 0);
#else
            *(float4*)ldst = *(const float4*)gsrc;
#endif
        }

        // ---- A tile (KT x MT) into LDS, k-major, transform fused ----
        if (TRANSA) {
            // rows are contiguous in m: 1024 float4 -> 4 per thread
            #pragma unroll
            for (int t = 0; t < (KT * MT) / (256 * 4); ++t) {
                int idx = tid + t * 256;
                int i  = idx >> 5;      // r offset (0..31)
                int j4 = idx & 31;      // float4 column in m
                float4 v = *(const float4*)&A[(long long)(r0 + i) * lda + (m0 + j4 * 4)];
                float s = scl[r0 + i];
                v.x = nan0(v.x) * s; v.y = nan0(v.y) * s;
                v.z = nan0(v.z) * s; v.w = nan0(v.w) * s;
                *(float4*)&ldsA[i][j4 * 4] = v;
            }
        } else {
            // rows are contiguous in r: vector load, scalar transposed store
            #pragma unroll
            for (int t = 0; t < (KT * MT) / (256 * 4); ++t) {
                int idx = tid + t * 256;
                int i  = idx >> 3;      // m offset (0..127)
                int j4 = idx & 7;       // float4 group along r
                float4 v = *(const float4*)&A[(long long)(m0 + i) * lda + (r0 + j4 * 4)];
                float vv[4] = {v.x, v.y, v.z, v.w};
                #pragma unroll
                for (int c = 0; c < 4; ++c) {
                    int rr = j4 * 4 + c;
                    float e = vv[c];
                    if (AMODE == 1)      e = nan0(e) * scl[m0 + i];
                    else if (AMODE == 2) e = (e - p0[r0 + rr]) * p1[r0 + rr] + p2[r0 + rr];
                    else if (AMODE == 3) e = silu(e);
                    ldsA[rr][i] = e;
                }
            }
        }

        // prefetch one line of the next A K-tile (global_prefetch_b8)
        if (r0 + KT < R) {
            if (TRANSA)
                __builtin_prefetch(&A[(long long)(r0 + KT + (tid & 31)) * lda + m0], 0, 1);
            else
                __builtin_prefetch(&A[(long long)(m0 + (tid & 127)) * lda + r0 + KT], 0, 1);
        }

        wait_async_lds();
        __syncthreads();

        // ---- 8 WMMA k-steps of 4, 2x2 16x16 tiles per wave ----
        #pragma unroll
        for (int kk = 0; kk < KT; kk += 4) {
            v2f af[2], bf[2];
            #pragma unroll
            for (int i = 0; i < 2; ++i) {
                int m = wm + i * 16 + lrow;
                af[i].x = ldsA[kk + 2 * half + 0][m];   // lanes<16: K0/K1, >=16: K2/K3
                af[i].y = ldsA[kk + 2 * half + 1][m];
            }
            #pragma unroll
            for (int j = 0; j < 2; ++j) {
                int n = wn + j * 16 + lrow;
                bf[j].x = ldsB[kk + 2 * half + 0][n];
                bf[j].y = ldsB[kk + 2 * half + 1][n];
            }
            #pragma unroll
            for (int i = 0; i < 2; ++i)
                #pragma unroll
                for (int j = 0; j < 2; ++j)
                    acc[i][j] = __builtin_amdgcn_wmma_f32_16x16x4_f32(
                        false, af[i], false, bf[j], (short)0, acc[i][j], false, false);
        }
        __syncthreads();
    }

    // ---- epilogue: C/D layout -> VGPR v: lanes 0-15 M=v, lanes 16-31 M=v+8 ----
    #pragma unroll
    for (int i = 0; i < 2; ++i)
        #pragma unroll
        for (int j = 0; j < 2; ++j) {
            int n = n0 + wn + j * 16 + lrow;
            float badd = HAS_BIAS ? bias[n] : 0.0f;
            #pragma unroll
            for (int v = 0; v < 8; ++v) {
                int m = m0 + wm + i * 16 + (half ? 8 : 0) + v;
                C[(long long)m * N + n] = acc[i][j][v] + badd;
            }
        }
}

// inv_norm[b,l] = 1/||nan0(EigVecs[b,l,:])||  (one block per row)
__global__ __launch_bounds__(256) void norm_kernel(
    const float* __restrict__ EigVecs, float* __restrict__ inv_norm, int K)
{
    __shared__ float red[256];
    const int row = blockIdx.x, tid = threadIdx.x;
    const float* q = EigVecs + (long long)row * K;
    float s = 0.0f;
    for (int k = tid * 4; k < K; k += 256 * 4) {
        float4 v = *(const float4*)&q[k];
        float a = nan0(v.x), b = nan0(v.y), c = nan0(v.z), d = nan0(v.w);
        s += a * a + b * b + c * c + d * d;
    }
    red[tid] = s; __syncthreads();
    for (int off = 128; off > 0; off >>= 1) {
        if (tid < off) red[tid] += red[tid + off];
        __syncthreads();
    }
    if (tid == 0) {
        float n = sqrtf(red[0]);
        inv_norm[row] = (n > 0.0f && isfinite(n)) ? (1.0f / n) : 0.0f;
    }
}

// Precompute: bn_g = gamma*rsqrt(var+eps); Abar = exp(-eps + delta*A) split re/im;
//             bc = delta*Bp*Cp  (all D*S = 2048 values)
__global__ void param_kernel(
    const float* __restrict__ log_Delta, const float* __restrict__ Bp,
    const float* __restrict__ Cp, const float* __restrict__ log_A_real,
    const float* __restrict__ A_imag, const float* __restrict__ bn_gamma,
    const float* __restrict__ bn_var,
    float* __restrict__ bn_g, float* __restrict__ abar_re,
    float* __restrict__ abar_im, float* __restrict__ bc)
{
    int idx = blockIdx.x * 256 + threadIdx.x;
    if (idx < 128) bn_g[idx] = bn_gamma[idx] * rsqrtf(bn_var[idx] + 1e-5f);
    if (idx < 2048) {
        int d = idx >> 4;
        float delta = __expf(log_Delta[d]);
        float a_re  = -__expf(log_A_real[idx]);
        float mag   = __expf(-1e-3f + delta * a_re);
        float th    = delta * A_imag[idx];
        abar_re[idx] = mag * __cosf(th);
        abar_im[idx] = mag * __sinf(th);
        bc[idx]      = delta * Bp[idx] * Cp[idx];
    }
}

// u = h[:, :128] * silu(h[:, 128:])
__global__ void gate_kernel(const float* __restrict__ h, float* __restrict__ u, int total)
{
    int idx = blockIdx.x * 256 + threadIdx.x;
    if (idx >= total) return;
    int row = idx >> 7, d = idx & 127;
    float x1 = h[(long long)row * 256 + d];
    float z  = h[(long long)row * 256 + 128 + d];
    u[idx] = x1 * silu(z);
}

// R[b,k,d] = QTX[b,k,d] * sum_s Re(z/(1-z)) * bc[d,s],  z = ev[b,k]*Abar[d,s]
__global__ __launch_bounds__(128) void wr_kernel(
    const float* __restrict__ EigVals, const float* __restrict__ qtx,
    const float* __restrict__ abar_re, const float* __restrict__ abar_im,
    const float* __restrict__ bc, float* __restrict__ r)
{
    const int bk = blockIdx.x;     // b*K + k
    const int d  = threadIdx.x;    // 0..127
    float ev = nan0(1.0f - EigVals[bk]);
    float w = 0.0f;
    #pragma unroll
    for (int s = 0; s < 16; ++s) {
        float ar = abar_re[d * 16 + s], ai = abar_im[d * 16 + s];
        float zr = ev * ar, zi = ev * ai;
        float den = (1.0f - zr) * (1.0f - zr) + zi * zi;
        float g = (zr - (zr * zr + zi * zi)) / den;   // Re(z/(1-z))
        w += g * bc[d * 16 + s];
    }
    r[(long long)bk * 128 + d] = qtx[(long long)bk * 128 + d] * w;
}

extern "C" void kernel_launch(void* const* d_in, const int* in_sizes, int n_in,
                              void* d_out, int out_size, void* d_ws, size_t ws_size,
                              hipStream_t stream)
{
    constexpr int B = 8, L = 1024, K = 1024, D = 128;
    const float* x         = (const float*)d_in[0];
    // d_in[1] = attn_mask (all ones, unused by the reference math)
    const float* EigVecs   = (const float*)d_in[2];
    const float* EigVals   = (const float*)d_in[3];
    const float* bn_gamma  = (const float*)d_in[4];
    const float* bn_beta   = (const float*)d_in[5];
    const float* bn_mean   = (const float*)d_in[6];
    const float* bn_var    = (const float*)d_in[7];
    const float* W_in      = (const float*)d_in[8];
    const float* b_in      = (const float*)d_in[9];
    const float* log_Delta = (const float*)d_in[10];
    const float* Bp        = (const float*)d_in[11];
    const float* Cp        = (const float*)d_in[12];
    const float* log_A_real= (const float*)d_in[13];
    const float* A_imag    = (const float*)d_in[14];
    const float* W_out     = (const float*)d_in[15];
    const float* b_out     = (const float*)d_in[16];
    float* out = (float*)d_out;

    // workspace layout (floats)
    float* ws       = (float*)d_ws;
    float* inv_norm = ws;                       // B*L            = 8192
    float* bn_g     = inv_norm + B * L;         // D              = 128
    float* abar_re  = bn_g + D;                 // D*S            = 2048
    float* abar_im  = abar_re + 2048;           // 2048
    float* bc       = abar_im + 2048;           // 2048
    float* h        = bc + 2048;                // B*L*2D         = 2097152
    float* u        = h + (long long)B * L * 2 * D;   // B*L*D    = 1048576
    float* qtx      = u + (long long)B * L * D;       // B*K*D    = 1048576
    float* rbuf     = qtx + (long long)B * K * D;     // B*K*D    = 1048576
    float* yssm     = rbuf + (long long)B * K * D;    // B*L*D    = 1048576

    // 1) pole/BN precompute + eigvec row norms
    param_kernel<<<8, 256, 0, stream>>>(log_Delta, Bp, Cp, log_A_real, A_imag,
                                        bn_gamma, bn_var, bn_g, abar_re, abar_im, bc);
    norm_kernel<<<B * L, 256, 0, stream>>>(EigVecs, inv_norm, K);

    // 2) in_proj: h = BN(x) @ W_in + b_in   (M=8192, N=256, R=128)
    wmma_gemm<2, false, true><<<dim3(64, 4, 1), 256, 0, stream>>>(
        x, W_in, h, bn_mean, bn_g, bn_beta, b_in,
        B * L, 2 * D, D, D, 0, 0, 0, 0);

    // 3) gated activation: u = h1 * silu(h2)
    gate_kernel<<<(B * L * D + 255) / 256, 256, 0, stream>>>(h, u, B * L * D);

    // 4) QTX[b] = (Q_norm)^T @ u   (per batch: M=K=1024, N=D=128, R=L=1024)
    wmma_gemm<1, true, false><<<dim3(K / MT, D / NT, B), 256, 0, stream>>>(
        EigVecs, u, qtx, inv_norm, inv_norm, inv_norm, b_in,
        K, D, L, K,
        (long long)L * K, (long long)L * D, (long long)K * D, (long long)L);

    // 5) SSM pole response: R = QTX * W
    wr_kernel<<<B * K, 128, 0, stream>>>(EigVals, qtx, abar_re, abar_im, bc, rbuf);

    // 6) yssm[b] = Q_norm @ R   (per batch: M=L=1024, N=D=128, R=K=1024)
    wmma_gemm<1, false, false><<<dim3(L / MT, D / NT, B), 256, 0, stream>>>(
        EigVecs, rbuf, yssm, inv_norm, inv_norm, inv_norm, b_in,
        L, D, K, K,
        (long long)L * K, (long long)K * D, (long long)L * D, (long long)L);

    // 7) out_proj: y = silu(yssm) @ W_out + b_out   (M=8192, N=128, R=128)
    wmma_gemm<3, false, true><<<dim3(64, 2, 1), 256, 0, stream>>>(
        yssm, W_out, out, bn_mean, bn_g, bn_beta, b_out,
        B * L, D, D, D, 0, 0, 0, 0);
}